// MemN2N_55954833933039
// MI455X (gfx1250) — compile-verified
//
#include <hip/hip_runtime.h>
#include <hip/hip_bf16.h>
#include <math.h>

#define BATCH 64
#define SSLOT 200
#define TOKS  50
#define VOC   100000
#define EMB   128
#define KHOP  3

typedef __attribute__((ext_vector_type(2))) float v2f;
typedef __attribute__((ext_vector_type(8))) float v8f;

// workspace layout (in floats):
//   bag[t] (t=0..3): [BATCH][SSLOT][EMB]   at t*BATCH*SSLOT*EMB
//   u:              [BATCH][EMB]           at 4*BATCH*SSLOT*EMB
#define WS_BAG(t) ((size_t)(t) * BATCH * SSLOT * EMB)
#define WS_U      ((size_t)4 * BATCH * SSLOT * EMB)

// ---------------------------------------------------------------------------
// Kernel 1: sentence embedding-bags for all 4 tables.
// One wave per sentence (lane holds float4 of the 128-dim embedding),
// 8 waves / 256 threads per block. grid = (BATCH*SSLOT/8, 4 tables).
// ---------------------------------------------------------------------------
__global__ void bag_kernel(const int* __restrict__ story,
                           const float* __restrict__ emb,
                           float* __restrict__ ws) {
  const int table = blockIdx.y;
  const int wave  = threadIdx.x >> 5;
  const int lane  = threadIdx.x & 31;
  const int sidx  = blockIdx.x * 8 + wave;          // 0 .. BATCH*SSLOT-1
  const int* toks = story + (size_t)sidx * TOKS;
  const float* tab = emb + (size_t)table * VOC * EMB;

  float4 acc = make_float4(0.f, 0.f, 0.f, 0.f);
  for (int t = 0; t < TOKS; ++t) {
    const int idx = toks[t];                        // uniform across wave
    const float4 v = *(const float4*)(tab + (size_t)idx * EMB + lane * 4);
    acc.x += v.x; acc.y += v.y; acc.z += v.z; acc.w += v.w;
  }
  *(float4*)(ws + WS_BAG(table) + (size_t)sidx * EMB + lane * 4) = acc;
}

// ---------------------------------------------------------------------------
// Kernel 2: question bag + 3 memory hops. One block (256 thr) per batch row.
// ---------------------------------------------------------------------------
__global__ void hops_kernel(const int* __restrict__ question,
                            const float* __restrict__ emb,
                            float* __restrict__ ws) {
  __shared__ float uL[EMB];
  __shared__ float probs[SSLOT];
  __shared__ float red[256];
  const int b    = blockIdx.x;
  const int tid  = threadIdx.x;
  const int lane = tid & 31;
  const int wave = tid >> 5;

  // u0 = sum-pooled question embedding under table 0
  if (tid < EMB) {
    float a = 0.f;
    const int* q = question + b * TOKS;
    for (int t = 0; t < TOKS; ++t) a += emb[(size_t)q[t] * EMB + tid];
    uL[tid] = a;
  }
  __syncthreads();

  for (int hop = 0; hop < KHOP; ++hop) {
    const float* bm = ws + WS_BAG(hop)     + (size_t)b * SSLOT * EMB;
    const float* bc = ws + WS_BAG(hop + 1) + (size_t)b * SSLOT * EMB;

    // scores[j] = dot(m[j], u); each wave handles 25 slots
    for (int jj = 0; jj < SSLOT / 8; ++jj) {
      const int j = wave * (SSLOT / 8) + jj;
      const float* row = bm + (size_t)j * EMB;
      float p = row[lane]      * uL[lane]
              + row[lane + 32] * uL[lane + 32]
              + row[lane + 64] * uL[lane + 64]
              + row[lane + 96] * uL[lane + 96];
      p += __shfl_down(p, 16, 32);
      p += __shfl_down(p,  8, 32);
      p += __shfl_down(p,  4, 32);
      p += __shfl_down(p,  2, 32);
      p += __shfl_down(p,  1, 32);
      if (lane == 0) probs[j] = p;
    }
    __syncthreads();

    // softmax over 200 slots
    red[tid] = (tid < SSLOT) ? probs[tid] : -3.0e38f;
    __syncthreads();
    for (int off = 128; off > 0; off >>= 1) {
      if (tid < off) red[tid] = fmaxf(red[tid], red[tid + off]);
      __syncthreads();
    }
    const float mx = red[0];
    __syncthreads();
    const float ex = (tid < SSLOT) ? __expf(probs[tid] - mx) : 0.f;
    red[tid] = ex;
    __syncthreads();
    for (int off = 128; off > 0; off >>= 1) {
      if (tid < off) red[tid] += red[tid + off];
      __syncthreads();
    }
    const float inv = 1.0f / red[0];
    if (tid < SSLOT) probs[tid] = ex * inv;
    __syncthreads();

    // u += c^T probs (residual)
    float nu = 0.f;
    if (tid < EMB) {
      nu = uL[tid];
      for (int j = 0; j < SSLOT; ++j) nu += probs[j] * bc[(size_t)j * EMB + tid];
    }
    __syncthreads();
    if (tid < EMB) uL[tid] = nu;
    __syncthreads();
  }
  if (tid < EMB) ws[WS_U + (size_t)b * EMB + tid] = uL[tid];
}

// ---------------------------------------------------------------------------
// Kernel 3: logits = u @ emb3^T via V_WMMA_F32_16X16X4_F32 (exact fp32).
// Block = 256 thr (8 waves) covers 64 batch x 32 vocab rows.
// LDS-staged tiles, padded stride 132 floats to avoid bank conflicts.
// ---------------------------------------------------------------------------
#define LDA 132
__global__ void logits_wmma_kernel(const float* __restrict__ emb,
                                   const float* __restrict__ uG,
                                   float* __restrict__ out) {
  __shared__ __align__(16) float ldsT[32 * LDA];     // vocab-row tile 32x128
  __shared__ __align__(16) float ldsU[BATCH * LDA];  // u: 64x128
  const int tid = threadIdx.x;
  const int v0  = blockIdx.x * 32;
  const float* tab = emb + (size_t)KHOP * VOC * EMB;

  // stage u (64x128) -> LDS
  for (int k = 0; k < 8; ++k) {
    const int id = tid + k * 256;                    // 0..2047 float4s
    const int r = id >> 5, q = id & 31;
    *(float4*)&ldsU[r * LDA + q * 4] = *(const float4*)&uG[r * EMB + q * 4];
  }
  // stage table tile (32x128) -> LDS
  for (int k = 0; k < 4; ++k) {
    const int id = tid + k * 256;                    // 0..1023 float4s
    const int r = id >> 5, q = id & 31;
    *(float4*)&ldsT[r * LDA + q * 4] =
        *(const float4*)&tab[(size_t)(v0 + r) * EMB + q * 4];
  }
  __syncthreads();

  const int lane = tid & 31, wave = tid >> 5;
  const int mt  = wave >> 1;        // batch tile 0..3 (M)
  const int nt  = wave & 1;         // vocab tile 0..1 (N)
  const int l16 = lane & 15;
  const int kh  = lane >> 4;        // selects K pair {0,1} vs {2,3}

  const float* Au = &ldsU[(mt * 16 + l16) * LDA + 2 * kh]; // A: M=batch
  const float* Bt = &ldsT[(nt * 16 + l16) * LDA + 2 * kh]; // B: N=vocab

  v8f acc = {};
  #pragma unroll
  for (int k0 = 0; k0 < EMB; k0 += 4) {
    v2f a  = *(const v2f*)(Au + k0);
    v2f bb = *(const v2f*)(Bt + k0);
    // D = A(16x4) x B(4x16) + C, fp32 exact
    acc = __builtin_amdgcn_wmma_f32_16x16x4_f32(false, a, false, bb,
                                                (short)0, acc, false, false);
  }

  const int brow0 = mt * 16 + 8 * kh;                // D: M rows split by lane half
  const int vcol  = v0 + nt * 16 + l16;              // N -> vocab (coalesced)
  #pragma unroll
  for (int r = 0; r < 8; ++r) {
    out[(size_t)(brow0 + r) * VOC + vcol] = acc[r];
  }
}

// ---------------------------------------------------------------------------
// Kernel 4: numerically-stable row softmax over 100000 classes, in place.
// One 1024-thread block per batch row; 3 grid-stride passes.
// ---------------------------------------------------------------------------
__global__ void softmax_rows_kernel(float* __restrict__ out) {
  __shared__ float red[1024];
  const int b = blockIdx.x, tid = threadIdx.x;
  float* row = out + (size_t)b * VOC;

  float mx = -3.0e38f;
  for (int i = tid; i < VOC; i += 1024) mx = fmaxf(mx, row[i]);
  red[tid] = mx; __syncthreads();
  for (int off = 512; off > 0; off >>= 1) {
    if (tid < off) red[tid] = fmaxf(red[tid], red[tid + off]);
    __syncthreads();
  }
  mx = red[0]; __syncthreads();

  float sum = 0.f;
  for (int i = tid; i < VOC; i += 1024) {
    const float e = __expf(row[i] - mx);
    row[i] = e;
    sum += e;
  }
  red[tid] = sum; __syncthreads();
  for (int off = 512; off > 0; off >>= 1) {
    if (tid < off) red[tid] += red[tid + off];
    __syncthreads();
  }
  const float inv = 1.0f / red[0];
  for (int i = tid; i < VOC; i += 1024) row[i] *= inv;
}

// ---------------------------------------------------------------------------
extern "C" void kernel_launch(void* const* d_in, const int* in_sizes, int n_in,
                              void* d_out, int out_size, void* d_ws, size_t ws_size,
                              hipStream_t stream) {
  (void)in_sizes; (void)n_in; (void)out_size; (void)ws_size;
  const int*   story    = (const int*)d_in[0];
  const int*   question = (const int*)d_in[1];
  const float* emb      = (const float*)d_in[2];
  float* out = (float*)d_out;
  float* ws  = (float*)d_ws;

  // 1) per-sentence embedding bags for all 4 tables (each table gathered once)
  bag_kernel<<<dim3(BATCH * SSLOT / 8, 4), 256, 0, stream>>>(story, emb, ws);
  // 2) question bag + 3 attention hops
  hops_kernel<<<BATCH, 256, 0, stream>>>(question, emb, ws);
  // 3) logits GEMM (M=64, N=100000, K=128) with f32 WMMA
  logits_wmma_kernel<<<VOC / 32, 256, 0, stream>>>(emb, ws + WS_U, out);
  // 4) stable softmax over vocab per batch row
  softmax_rows_kernel<<<BATCH, 1024, 0, stream>>>(out);
}